// BatchedGraphAttentionLayer_47605417509428
// MI455X (gfx1250) — compile-verified
//
#include <hip/hip_runtime.h>
#include <math.h>

typedef float v2f __attribute__((ext_vector_type(2)));
typedef float v8f __attribute__((ext_vector_type(8)));

#define B_ 32
#define S_ 20
#define N_ 128
#define F_ 64
#define NH 4
#define ND 64
#define O_ 256  /* NH*ND */

/* LDS strides (floats) */
#define HTT_STRIDE 136   /* ht^T: [256 feats][128 nodes + pad], 544B rows */
#define W_STRIDE   68
#define H_STRIDE   68
#define ATT_STRIDE 132
#define OUT_STRIDE 68

/* LDS layout (float offsets). sATT reuses sW space, sOUT reuses sH space. */
#define OFF_HTT  0
#define SZ_HTT   (O_ * HTT_STRIDE)         /* 34816 */
#define OFF_SRC  (OFF_HTT + SZ_HTT)        /* 34816 */
#define OFF_TGT  (OFF_SRC + NH * N_)       /* 35328 */
#define OFF_MASK (OFF_TGT + NH * N_)       /* 35840: 128 rows x 4 u32 words */
#define OFF_W    (OFF_MASK + N_ * 4)       /* 36352 */
#define SZ_W     (O_ * W_STRIDE)           /* 17408 */
#define OFF_H    (OFF_W + SZ_W)            /* 53760 */
#define SZ_H     (N_ * H_STRIDE)           /* 8704  */
#define LDS_FLOATS (OFF_H + SZ_H)          /* 62464 floats = 249856 B <= 320KB */
#define OFF_ATT  OFF_W                     /* 128*132 = 16896 <= 17408 */
#define OFF_OUT  OFF_H

__global__ __launch_bounds__(256) void gat_bsn_kernel(
    const float* __restrict__ hgl, const float* __restrict__ adj,
    const float* __restrict__ Wgl, const float* __restrict__ agl,
    float* __restrict__ outgl)
{
  extern __shared__ float lds[];
  const int bs   = blockIdx.x;       /* one (b,s) per workgroup: 640 blocks */
  const int tid  = threadIdx.x;
  const int wave = tid >> 5;
  const int lane = tid & 31;
  const int l16  = lane & 15;
  const int hi   = lane >> 4;        /* 0: K=0/1 half, 1: K=2/3 half */

  const float* hb   = hgl   + (size_t)bs * N_ * F_;
  const float* adjb = adj   + (size_t)bs * N_ * N_;
  float*       outb = outgl + (size_t)bs * N_ * F_;

  float*    sHTT  = lds + OFF_HTT;
  float*    sSRC  = lds + OFF_SRC;
  float*    sTGT  = lds + OFF_TGT;
  unsigned* sMASK = (unsigned*)(lds + OFF_MASK);
  float*    sW    = lds + OFF_W;
  float*    sH    = lds + OFF_H;
  float*    sATT  = lds + OFF_ATT;
  float*    sOUT  = lds + OFF_OUT;

  /* ---- Phase 0: stage h (stride 68), W^T (256x64), adj bitmask ---- */
  for (int i = tid; i < N_ * F_ / 4; i += 256) {   /* h: float4 coalesced */
    int n  = i >> 4;
    int f4 = i & 15;
    float4 v = ((const float4*)hb)[i];
    *(float4*)&sH[n * H_STRIDE + f4 * 4] = v;
  }
  for (int i = tid; i < F_ * O_ / 4; i += 256) {   /* W: [64][256] row-major */
    float4 v = ((const float4*)Wgl)[i];
    int f = (i * 4) >> 8;
    int o = (i * 4) & 255;
    sW[(o + 0) * W_STRIDE + f] = v.x;              /* scatter-transpose to W^T */
    sW[(o + 1) * W_STRIDE + f] = v.y;
    sW[(o + 2) * W_STRIDE + f] = v.z;
    sW[(o + 3) * W_STRIDE + f] = v.w;
  }
  /* adj predicate -> 1 bit; adj crosses HBM exactly once. 512 u32 words. */
  for (int i = tid; i < N_ * 4; i += 256) {
    int rowm = i >> 2;
    int seg  = i & 3;
    const float4* ar = (const float4*)(adjb + rowm * N_ + seg * 32);
    unsigned m = 0;
    #pragma unroll
    for (int q = 0; q < 8; ++q) {
      float4 v = ar[q];
      m |= (v.x < 0.05f ? 1u : 0u) << (q * 4 + 0);
      m |= (v.y < 0.05f ? 1u : 0u) << (q * 4 + 1);
      m |= (v.z < 0.05f ? 1u : 0u) << (q * 4 + 2);
      m |= (v.w < 0.05f ? 1u : 0u) << (q * 4 + 3);
    }
    sMASK[i] = m;
  }
  __syncthreads();

  /* ---- Phase 1: ht = h @ W (128x256), f32 WMMA 16x16x4; D stored transposed.
     128 tiles of 16x16; uniform trip count: wave w owns tiles w, w+8, ... */
  #pragma unroll 1
  for (int it = 0; it < 16; ++it) {
    int t  = (it << 3) + wave;
    int mi = t >> 4;                 /* 0..7  node-tile  */
    int oi = t & 15;                 /* 0..15 feat-tile  */
    v8f acc = {};
    int m = mi * 16 + l16;
    int o = oi * 16 + l16;
    for (int k0 = 0; k0 < F_; k0 += 4) {
      int kk = k0 + (hi << 1);
      v2f Af = *(const v2f*)&sH[m * H_STRIDE + kk];   /* A[m][kk..kk+1]      */
      v2f Bf = *(const v2f*)&sW[o * W_STRIDE + kk];   /* W^T[o][kk]=B[kk][o] */
      acc = __builtin_amdgcn_wmma_f32_16x16x4_f32(
          false, Af, false, Bf, (short)0, acc, false, false);
    }
    /* D element (row=node mbase+r, col=feat o) -> sHTT[o][node]; 8 consecutive
       floats per lane -> two 16B stores. */
    int mbase = mi * 16 + (hi << 3);
    float* dst = &sHTT[(oi * 16 + l16) * HTT_STRIDE + mbase];
    *(float4*)&dst[0] = make_float4(acc[0], acc[1], acc[2], acc[3]);
    *(float4*)&dst[4] = make_float4(acc[4], acc[5], acc[6], acc[7]);
  }
  __syncthreads();

  /* ---- Phase 2: src/tgt head projections; zero output accumulator ---- */
  for (int i = tid; i < SZ_H; i += 256) sOUT[i] = 0.0f;  /* reuses sH space */
  for (int i = tid; i < NH * N_; i += 256) {
    int hh = i >> 7;
    int n  = i & 127;
    const float* arow = agl + hh * 2 * ND;  /* a[h][0:64]=src, [64:128]=tgt */
    float s = 0.0f, t = 0.0f;
    for (int d = 0; d < ND; ++d) {
      float v = sHTT[(hh * ND + d) * HTT_STRIDE + n];  /* lanes: n+1 apart */
      s += v * arow[d];
      t += v * arow[ND + d];
    }
    sSRC[i] = s;
    sTGT[i] = t;
  }
  __syncthreads();

  /* ---- Phase 3: per-head masked-sigmoid softmax + attn @ ht_head ---- */
  const int row  = tid >> 1;         /* 2 threads per attention row */
  const int half = tid & 1;
  for (int hh = 0; hh < NH; ++hh) {
    /* 3a: logits (sigmoid or MASK_FILL) from LDS only, then row softmax. */
    float si = sSRC[hh * N_ + row];
    unsigned mw0 = sMASK[row * 4 + half * 2 + 0];
    unsigned mw1 = sMASK[row * 4 + half * 2 + 1];
    float* attrow = &sATT[row * ATT_STRIDE + half * 64];
    float mx = -3.0e38f;
    for (int j = 0; j < 64; ++j) {
      unsigned mw = (j < 32) ? mw0 : mw1;
      float tj = sTGT[hh * N_ + half * 64 + j];
      float e  = 1.0f / (1.0f + __expf(-(si + tj)));
      if ((mw >> (j & 31)) & 1u) e = -9.0e15f;   /* MASK_FILL */
      attrow[j] = e;
      mx = fmaxf(mx, e);
    }
    mx = fmaxf(mx, __shfl_xor(mx, 1, 32));
    float sum = 0.0f;
    for (int j = 0; j < 64; ++j) {
      float e = __expf(attrow[j] - mx);
      attrow[j] = e;
      sum += e;
    }
    sum += __shfl_xor(sum, 1, 32);
    float inv = 1.0f / sum;
    for (int j = 0; j < 64; ++j) attrow[j] *= inv;
    __syncthreads();

    /* 3b: out_h = attn(128x128) @ ht_head(128x64); 32 tiles, 4 per wave.
       B fragments are contiguous float2 in ht^T. */
    #pragma unroll 1
    for (int it = 0; it < 4; ++it) {
      int t  = (it << 3) + wave;
      int mi = t >> 2;               /* 0..7 node-tile */
      int ni = t & 3;                /* 0..3 feat-tile */
      v8f acc = {};
      int m = mi * 16 + l16;
      int n = ni * 16 + l16;
      const float* brow = &sHTT[(hh * ND + n) * HTT_STRIDE];
      for (int k0 = 0; k0 < N_; k0 += 4) {
        int kk = k0 + (hi << 1);
        v2f Af = *(const v2f*)&sATT[m * ATT_STRIDE + kk]; /* A[m][kk..kk+1] */
        v2f Bf = *(const v2f*)&brow[kk];                  /* B[kk..kk+1][n] */
        acc = __builtin_amdgcn_wmma_f32_16x16x4_f32(
            false, Af, false, Bf, (short)0, acc, false, false);
      }
      int mbase = mi * 16 + (hi << 3);
      #pragma unroll
      for (int r = 0; r < 8; ++r)
        sOUT[(mbase + r) * OUT_STRIDE + ni * 16 + l16] += acc[r] * 0.25f;
    }
    __syncthreads();
  }

  /* ---- Phase 4: write output (B,S,N,64), coalesced ---- */
  for (int i = tid; i < N_ * F_; i += 256) {
    int n = i >> 6;
    int f = i & 63;
    outb[i] = sOUT[n * OUT_STRIDE + f];
  }
}

extern "C" void kernel_launch(void* const* d_in, const int* in_sizes, int n_in,
                              void* d_out, int out_size, void* d_ws, size_t ws_size,
                              hipStream_t stream) {
  (void)in_sizes; (void)n_in; (void)out_size; (void)d_ws; (void)ws_size;
  const float* h   = (const float*)d_in[0];
  const float* adj = (const float*)d_in[1];
  const float* W   = (const float*)d_in[2];
  const float* a   = (const float*)d_in[3];
  float* out = (float*)d_out;

  const size_t lds_bytes = (size_t)LDS_FLOATS * sizeof(float); /* 249856 B */
  hipFuncSetAttribute((const void*)gat_bsn_kernel,
                      hipFuncAttributeMaxDynamicSharedMemorySize,
                      (int)lds_bytes);
  hipLaunchKernelGGL(gat_bsn_kernel, dim3(B_ * S_), dim3(256), lds_bytes, stream,
                     h, adj, W, a, out);
}